// SimpleSAGE_64587718197583
// MI455X (gfx1250) — compile-verified
//
#include <hip/hip_runtime.h>

typedef __attribute__((ext_vector_type(2))) float v2f;
typedef __attribute__((ext_vector_type(8))) float v8f;

#define N_NODES 100000
#define N_EDGES 3200000

// ---------------- utility kernels ----------------

__global__ void fill_zero_f(float* __restrict__ p, int n) {
  int i = blockIdx.x * blockDim.x + threadIdx.x;
  if (i < n) p[i] = 0.0f;
}

__global__ void degree_kernel(const long long* __restrict__ dst, float* __restrict__ deg, int ne) {
  int e = blockIdx.x * blockDim.x + threadIdx.x;
  if (e < ne) atomicAdd(&deg[(int)dst[e]], 1.0f);
}

__global__ void invert_deg_kernel(float* __restrict__ deg, int n) {
  int i = blockIdx.x * blockDim.x + threadIdx.x;
  if (i < n) deg[i] = 1.0f / fmaxf(deg[i], 1.0f);
}

// pad x [N,13] -> [N,16] (channels 13..15 = 0)
__global__ void pad_x_kernel(const float* __restrict__ x, float* __restrict__ xp, int n) {
  int i = blockIdx.x * blockDim.x + threadIdx.x;
  if (i < n * 16) {
    int node = i >> 4, c = i & 15;
    xp[i] = (c < 13) ? x[node * 13 + c] : 0.0f;
  }
}

// pad layer-0 weights [13,64] -> [16,64] (rows 13..15 = 0)
__global__ void pad_w0_kernel(const float* __restrict__ wl, const float* __restrict__ wr,
                              float* __restrict__ wlp, float* __restrict__ wrp) {
  int i = blockIdx.x * blockDim.x + threadIdx.x;
  if (i < 16 * 64) {
    int k = i >> 6;
    wlp[i] = (k < 13) ? wl[i] : 0.0f;
    wrp[i] = (k < 13) ? wr[i] : 0.0f;
  }
}

// ---------------- edge scatter (segment_sum) ----------------
// 16-thread groups per edge (d=64) / 4-thread groups (d=16): float4 gather of
// the 256B source row is fully coalesced; scatter via 4 float atomics to the
// L2-resident accumulator.
__global__ void scatter_add_kernel(const float* __restrict__ h, const long long* __restrict__ src,
                                   const long long* __restrict__ dst, float* __restrict__ agg,
                                   int n_edges, int d) {
  int tid = blockIdx.x * blockDim.x + threadIdx.x;
  int dq = d >> 2;
  if (tid >= n_edges * dq) return;
  int e = tid / dq;
  int c = (tid - e * dq) << 2;
  int s = (int)src[e];
  int t = (int)dst[e];
  const float4 v = *(const float4*)(h + (size_t)s * d + c);
  float* a = agg + (size_t)t * d + c;
  atomicAdd(a + 0, v.x);
  atomicAdd(a + 1, v.y);
  atomicAdd(a + 2, v.z);
  atomicAdd(a + 3, v.w);
}

// ---------------- dense layer: out = (A*inv_deg)@Wl + X@Wr + b, opt ReLU ----
// One wave per 16x16 output tile. f32 WMMA 16x16x4, K-loop step 4.
// A-frag layout (16x4 f32): lanes 0-15 hold M=lane, {K=k,k+1}; lanes 16-31 hold
// M=lane-16, {K=k+2,k+3}. B-frag mirrors with N on lanes. C/D: vgpr j =
// D[m0 + half*8 + j][n0 + lane%16].
__global__ void sage_gemm_wmma(const float* __restrict__ A, const float* __restrict__ X,
                               const float* __restrict__ inv_deg,
                               const float* __restrict__ Wl, const float* __restrict__ Wr,
                               const float* __restrict__ bias, float* __restrict__ out,
                               int K, int dout, int relu, int num_tiles) {
  int wave = (blockIdx.x * blockDim.x + threadIdx.x) >> 5;
  if (wave >= num_tiles) return;          // wave-uniform: EXEC all-ones for WMMA
  int lane = threadIdx.x & 31;
  int tpr = dout >> 4;
  int m0 = (wave / tpr) << 4;
  int n0 = (wave % tpr) << 4;
  int half = lane >> 4;
  int r = lane & 15;

  v8f c = {};

  // pass 1: mean-neighbor path, inv_deg folded into A fragment
  {
    const float* Ar = A + (size_t)(m0 + r) * K;
    float s = inv_deg[m0 + r];
    for (int k = 0; k < K; k += 4) {
      v2f a, b;
      a[0] = Ar[k + 2 * half] * s;
      a[1] = Ar[k + 2 * half + 1] * s;
      b[0] = Wl[(size_t)(k + 2 * half) * dout + n0 + r];
      b[1] = Wl[(size_t)(k + 2 * half + 1) * dout + n0 + r];
      c = __builtin_amdgcn_wmma_f32_16x16x4_f32(false, a, false, b, (short)0, c, false, false);
    }
  }
  // pass 2: root path
  {
    const float* Xr = X + (size_t)(m0 + r) * K;
    for (int k = 0; k < K; k += 4) {
      v2f a, b;
      a[0] = Xr[k + 2 * half];
      a[1] = Xr[k + 2 * half + 1];
      b[0] = Wr[(size_t)(k + 2 * half) * dout + n0 + r];
      b[1] = Wr[(size_t)(k + 2 * half + 1) * dout + n0 + r];
      c = __builtin_amdgcn_wmma_f32_16x16x4_f32(false, a, false, b, (short)0, c, false, false);
    }
  }

  int col = n0 + r;
  float bv = bias[col];
  int rbase = m0 + half * 8;
#pragma unroll
  for (int j = 0; j < 8; ++j) {
    float v = c[j] + bv;
    if (relu) v = fmaxf(v, 0.0f);
    out[(size_t)(rbase + j) * dout + col] = v;
  }
}

// ---------------- final layer (64 -> 1): wave per node ----------------
__global__ void sage_final_kernel(const float* __restrict__ agg, const float* __restrict__ h,
                                  const float* __restrict__ inv_deg,
                                  const float* __restrict__ Wl, const float* __restrict__ Wr,
                                  const float* __restrict__ b, float* __restrict__ out,
                                  int n_nodes) {
  int wave = (blockIdx.x * blockDim.x + threadIdx.x) >> 5;
  int lane = threadIdx.x & 31;
  if (wave >= n_nodes) return;
  float s = inv_deg[wave];
  const float* ar = agg + (size_t)wave * 64;
  const float* hr = h + (size_t)wave * 64;
  float acc = ar[lane] * s * Wl[lane] + ar[lane + 32] * s * Wl[lane + 32]
            + hr[lane] * Wr[lane] + hr[lane + 32] * Wr[lane + 32];
#pragma unroll
  for (int off = 16; off > 0; off >>= 1) acc += __shfl_xor(acc, off, 32);
  if (lane == 0) out[wave] = acc + b[0];
}

// ---------------- host orchestration ----------------

extern "C" void kernel_launch(void* const* d_in, const int* in_sizes, int n_in,
                              void* d_out, int out_size, void* d_ws, size_t ws_size,
                              hipStream_t stream) {
  const float* x       = (const float*)d_in[0];
  const long long* ei  = (const long long*)d_in[1];
  const long long* src = ei;
  const long long* dst = ei + N_EDGES;
  const float* Wl0 = (const float*)d_in[2];
  const float* bl0 = (const float*)d_in[3];
  const float* Wr0 = (const float*)d_in[4];
  const float* Wl1 = (const float*)d_in[5];
  const float* bl1 = (const float*)d_in[6];
  const float* Wr1 = (const float*)d_in[7];
  const float* Wl2 = (const float*)d_in[8];
  const float* bl2 = (const float*)d_in[9];
  const float* Wr2 = (const float*)d_in[10];
  const float* Wl3 = (const float*)d_in[11];
  const float* bl3 = (const float*)d_in[12];
  const float* Wr3 = (const float*)d_in[13];
  float* out = (float*)d_out;

  // workspace carve-out (256B aligned)
  char* wp = (char*)d_ws;
  auto carve = [&](size_t bytes) -> float* {
    float* p = (float*)wp;
    wp += (bytes + 255) & ~(size_t)255;
    return p;
  };
  float* deg  = carve((size_t)N_NODES * 4);            // 0.4 MB (becomes inv_deg)
  float* xpad = carve((size_t)N_NODES * 16 * 4);       // 6.4 MB
  float* agg  = carve((size_t)N_NODES * 64 * 4);       // 25.6 MB
  float* hA   = carve((size_t)N_NODES * 64 * 4);       // 25.6 MB
  float* hB   = carve((size_t)N_NODES * 64 * 4);       // 25.6 MB
  float* w0lp = carve(16 * 64 * 4);
  float* w0rp = carve(16 * 64 * 4);

  const int T = 256;
  const int tiles = (N_NODES / 16) * (64 / 16);        // 25000, divisible by 8

  // degrees -> inverse degrees
  fill_zero_f<<<(N_NODES + T - 1) / T, T, 0, stream>>>(deg, N_NODES);
  degree_kernel<<<(N_EDGES + T - 1) / T, T, 0, stream>>>(dst, deg, N_EDGES);
  invert_deg_kernel<<<(N_NODES + T - 1) / T, T, 0, stream>>>(deg, N_NODES);

  // padded inputs/weights for layer 0 (K: 13 -> 16)
  pad_x_kernel<<<(N_NODES * 16 + T - 1) / T, T, 0, stream>>>(x, xpad, N_NODES);
  pad_w0_kernel<<<4, T, 0, stream>>>(Wl0, Wr0, w0lp, w0rp);

  // ---- layer 0: xpad[N,16] -> hA[N,64]
  fill_zero_f<<<(N_NODES * 16 + T - 1) / T, T, 0, stream>>>(agg, N_NODES * 16);
  scatter_add_kernel<<<(unsigned)((N_EDGES * 4 + T - 1) / T), T, 0, stream>>>(
      xpad, src, dst, agg, N_EDGES, 16);
  sage_gemm_wmma<<<(tiles * 32 + T - 1) / T, T, 0, stream>>>(
      agg, xpad, deg, w0lp, w0rp, bl0, hA, 16, 64, 1, tiles);

  // ---- layer 1: hA -> hB
  fill_zero_f<<<(N_NODES * 64 + T - 1) / T, T, 0, stream>>>(agg, N_NODES * 64);
  scatter_add_kernel<<<(unsigned)(((size_t)N_EDGES * 16 + T - 1) / T), T, 0, stream>>>(
      hA, src, dst, agg, N_EDGES, 64);
  sage_gemm_wmma<<<(tiles * 32 + T - 1) / T, T, 0, stream>>>(
      agg, hA, deg, Wl1, Wr1, bl1, hB, 64, 64, 1, tiles);

  // ---- layer 2: hB -> hA
  fill_zero_f<<<(N_NODES * 64 + T - 1) / T, T, 0, stream>>>(agg, N_NODES * 64);
  scatter_add_kernel<<<(unsigned)(((size_t)N_EDGES * 16 + T - 1) / T), T, 0, stream>>>(
      hB, src, dst, agg, N_EDGES, 64);
  sage_gemm_wmma<<<(tiles * 32 + T - 1) / T, T, 0, stream>>>(
      agg, hB, deg, Wl2, Wr2, bl2, hA, 64, 64, 1, tiles);

  // ---- layer 3: hA -> out[N] (d_out = 1)
  fill_zero_f<<<(N_NODES * 64 + T - 1) / T, T, 0, stream>>>(agg, N_NODES * 64);
  scatter_add_kernel<<<(unsigned)(((size_t)N_EDGES * 16 + T - 1) / T), T, 0, stream>>>(
      hA, src, dst, agg, N_EDGES, 64);
  sage_final_kernel<<<(N_NODES * 32 + T - 1) / T, T, 0, stream>>>(
      agg, hA, deg, Wl3, Wr3, bl3, out, N_NODES);
}